// ImagePlanes_28381143892676
// MI455X (gfx1250) — compile-verified
//
#include <hip/hip_runtime.h>
#include <hip/hip_bf16.h>

// ---------------------------------------------------------------------------
// MI455X (gfx1250) ImagePlanes kernel.
//
// Roofline: output = 323 MB streamed writes -> ~14us floor at 23.3 TB/s HBM.
// Images (30.7 MB) live in the 192 MB L2, so bilinear gathers are L2-resident.
// => nontemporal wide stores for output, WMMA f32 16x16x4 for the 48x4 @ 4xP
//    projection GEMM, wave32 tile of 16 points per wave.
// ---------------------------------------------------------------------------

#define FOCALC   555.555f
#define IMG      400
#define PADC     5
#define NVIEWS   16
#define NPOINTS  65536
#define FEATS_COUNT (NPOINTS * NVIEWS * 75)   // 78,643,200 floats

typedef float v2f __attribute__((ext_vector_type(2)));
typedef float v4f __attribute__((ext_vector_type(4)));
typedef float v8f __attribute__((ext_vector_type(8)));
typedef v4f v4f_u __attribute__((aligned(4)));   // unaligned 128b store type

__device__ __forceinline__ int iclampi(int v, int lo, int hi) {
    return v < lo ? lo : (v > hi ? hi : v);
}

// ---------------------------------------------------------------------------
// Prep: A[48][4] = K @ inv(pose @ diag(-1,1,1,1))[:3,:]  per view.
// Rotation part of pose is orthonormal (built that way), so the inverse of
// the rigid transform is [R'^T | -R'^T t] exactly.
// ---------------------------------------------------------------------------
__global__ void ip_prep(const float* __restrict__ poses, float* __restrict__ A) {
    int v = threadIdx.x;
    if (v >= NVIEWS) return;
    const float* P = poses + v * 16;
    float R[3][3];
#pragma unroll
    for (int r = 0; r < 3; ++r) {
        R[r][0] = -P[r * 4 + 0];   // flip negates column 0
        R[r][1] =  P[r * 4 + 1];
        R[r][2] =  P[r * 4 + 2];
    }
    float t0 = P[3], t1 = P[7], t2 = P[11];
    float M[3][4];
#pragma unroll
    for (int i = 0; i < 3; ++i) {
        M[i][0] = R[0][i];
        M[i][1] = R[1][i];
        M[i][2] = R[2][i];
        M[i][3] = -(R[0][i] * t0 + R[1][i] * t1 + R[2][i] * t2);
    }
#pragma unroll
    for (int j = 0; j < 4; ++j) {
        A[(3 * v + 0) * 4 + j] = FOCALC * M[0][j] + 200.0f * M[2][j]; // K row0
        A[(3 * v + 1) * 4 + j] = FOCALC * M[1][j] + 200.0f * M[2][j]; // K row1
        A[(3 * v + 2) * 4 + j] = M[2][j];                             // K row2
    }
}

// ---------------------------------------------------------------------------
// Main kernel: 512 blocks x 256 threads (8 waves). Each wave owns 16 points.
//   Stage 1: project its 16 points into all 16 views via 3x
//            V_WMMA_F32_16X16X4_F32 (48x4 A-stack  x  4x16 points).
//   Stage 2: scatter D through LDS (same-wave, in-order), then each lane
//            handles one (point, view) pair per iteration: 5x5 bilinear crop
//            + nontemporal b128 stores of 75 floats.
// ---------------------------------------------------------------------------
__launch_bounds__(256, 2)
__global__ void ip_main(const float* __restrict__ points,
                        const float* __restrict__ images,
                        const float* __restrict__ Amat,
                        float* __restrict__ feats,
                        float* __restrict__ coord) {
    __shared__ __align__(16) float ps_s[8][16][48];   // [wave][point][48 rows]

    const int  lane = threadIdx.x & 31;
    const int  wv   = threadIdx.x >> 5;
    const bool hihalf = lane >= 16;
    const int  n    = lane & 15;                       // point-in-tile / view id
    const int  tileBase = blockIdx.x * 128 + wv * 16;

    // --- A operand: lane<16 holds K=0,1 of row n; lane>=16 holds K=2,3 (ISA 7.12.2)
    const int colbase = hihalf ? 2 : 0;
    v2f a0 = *(const v2f*)(Amat + (0 * 16 + n) * 4 + colbase);
    v2f a1 = *(const v2f*)(Amat + (1 * 16 + n) * 4 + colbase);
    v2f a2 = *(const v2f*)(Amat + (2 * 16 + n) * 4 + colbase);

    // --- B operand: column n = homogeneous point; lanes<16 K=0,1 (x,y),
    //     lanes>=16 K=2,3 (z,1). Branchless so EXEC stays all-ones for WMMA.
    const float* pp = points + (size_t)(tileBase + n) * 3;
    float px = pp[0], py = pp[1], pz = pp[2];
    v2f b;
    b.x = hihalf ? pz : px;
    b.y = hihalf ? 1.0f : py;

    v8f cacc = {0.f, 0.f, 0.f, 0.f, 0.f, 0.f, 0.f, 0.f};
    v8f d0 = __builtin_amdgcn_wmma_f32_16x16x4_f32(false, a0, false, b, (short)0, cacc, false, false);
    v8f d1 = __builtin_amdgcn_wmma_f32_16x16x4_f32(false, a1, false, b, (short)0, cacc, false, false);
    v8f d2 = __builtin_amdgcn_wmma_f32_16x16x4_f32(false, a2, false, b, (short)0, cacc, false, false);

    // --- Scatter D to LDS as ps_s[point][row48]: VGPR r holds M=r (lanes 0-15)
    //     or M=r+8 (lanes 16-31); per-lane the 8 rows are contiguous -> b128 x2.
    const int roff = hihalf ? 8 : 0;
    float* row = &ps_s[wv][n][0];
    *(v4f*)(row + roff + 0)      = (v4f){d0[0], d0[1], d0[2], d0[3]};
    *(v4f*)(row + roff + 4)      = (v4f){d0[4], d0[5], d0[6], d0[7]};
    *(v4f*)(row + 16 + roff + 0) = (v4f){d1[0], d1[1], d1[2], d1[3]};
    *(v4f*)(row + 16 + roff + 4) = (v4f){d1[4], d1[5], d1[6], d1[7]};
    *(v4f*)(row + 32 + roff + 0) = (v4f){d2[0], d2[1], d2[2], d2[3]};
    *(v4f*)(row + 32 + roff + 4) = (v4f){d2[4], d2[5], d2[6], d2[7]};

    // Same-wave LDS ops are in-order; this wait + compiler barrier makes the
    // cross-lane redistribution explicit (no workgroup barrier needed).
    asm volatile("s_wait_dscnt 0" ::: "memory");

    const float SCALE = 409.0f / 410.0f;             // (Hp-1)/Hp, Hp=410
    const int   view  = n;                            // lane's fixed view
    const float* imgv = images + (size_t)view * (IMG * IMG * 3);

#pragma unroll 1
    for (int k = 0; k < 8; ++k) {
        const int p_local = (k << 1) | (hihalf ? 1 : 0);
        const int p       = tileBase + p_local;

        float psx = ps_s[wv][p_local][3 * view + 0];
        float psy = ps_s[wv][p_local][3 * view + 1];
        float psw = ps_s[wv][p_local][3 * view + 2];

        float pu = fminf(fmaxf((psx / psw) * (1.0f / IMG), 0.0f), 1.0f);
        float pv = fminf(fmaxf((psy / psw) * (1.0f / IMG), 0.0f), 1.0f);

        v2f cn = {2.0f * pu - 1.0f, 2.0f * pv - 1.0f};
        __builtin_nontemporal_store(cn, (v2f*)(coord + ((size_t)view * NPOINTS + p) * 2));

        // pixel coords + PADDING - CROP_STEP
        float x = pu * IMG + 3.0f;
        float y = pv * IMG + 3.0f;

        // column corners (depend only on j): clamp address, mask value (zero-pad)
        float wc[5], mcA[5], mcB[5];
        int   colA[5], colB[5];
#pragma unroll
        for (int j = 0; j < 5; ++j) {
            float cj  = (y + (float)j) * SCALE;
            float c0f = floorf(cj);
            wc[j] = cj - c0f;
            int ci = (int)c0f - PADC;                 // padded -> original coords
            colA[j] = iclampi(ci, 0, IMG - 1) * 3;
            mcA[j]  = (ci >= 0 && ci < IMG) ? 1.0f : 0.0f;
            int ci1 = ci + 1;
            colB[j] = iclampi(ci1, 0, IMG - 1) * 3;
            mcB[j]  = (ci1 >= 0 && ci1 < IMG) ? 1.0f : 0.0f;
        }

        float out[75];
#pragma unroll
        for (int i = 0; i < 5; ++i) {
            float rif = (x + (float)i) * SCALE;
            float r0f = floorf(rif);
            float wr  = rif - r0f;
            int ro  = (int)r0f - PADC;
            int rowA = iclampi(ro, 0, IMG - 1) * (IMG * 3);
            float mrA = (ro >= 0 && ro < IMG) ? 1.0f : 0.0f;
            int ro1 = ro + 1;
            int rowB = iclampi(ro1, 0, IMG - 1) * (IMG * 3);
            float mrB = (ro1 >= 0 && ro1 < IMG) ? 1.0f : 0.0f;
#pragma unroll
            for (int j = 0; j < 5; ++j) {
                float w00 = (1.0f - wr) * (1.0f - wc[j]);
                float w01 = (1.0f - wr) * wc[j];
                float w10 = wr * (1.0f - wc[j]);
                float w11 = wr * wc[j];
                float g00 = w00 * (mrA * mcA[j]);
                float g01 = w01 * (mrA * mcB[j]);
                float g10 = w10 * (mrB * mcA[j]);
                float g11 = w11 * (mrB * mcB[j]);
                const float* p00 = imgv + rowA + colA[j];
                const float* p01 = imgv + rowA + colB[j];
                const float* p10 = imgv + rowB + colA[j];
                const float* p11 = imgv + rowB + colB[j];
#pragma unroll
                for (int ch = 0; ch < 3; ++ch) {
                    out[ch * 25 + i * 5 + j] =
                        p00[ch] * g00 + p01[ch] * g01 + p10[ch] * g10 + p11[ch] * g11;
                }
            }
        }

        // 300 bytes per (point,view): 18x NT b128 (4-byte aligned) + 3x NT b32.
        float* dst = feats + ((size_t)p * NVIEWS + view) * 75;
#pragma unroll
        for (int q = 0; q < 18; ++q) {
            v4f val = {out[q * 4 + 0], out[q * 4 + 1], out[q * 4 + 2], out[q * 4 + 3]};
            __builtin_nontemporal_store(val, (v4f_u*)(dst + q * 4));
        }
        __builtin_nontemporal_store(out[72], dst + 72);
        __builtin_nontemporal_store(out[73], dst + 73);
        __builtin_nontemporal_store(out[74], dst + 74);
    }
}

extern "C" void kernel_launch(void* const* d_in, const int* in_sizes, int n_in,
                              void* d_out, int out_size, void* d_ws, size_t ws_size,
                              hipStream_t stream) {
    const float* points = (const float*)d_in[0];   // (65536, 3)
    const float* images = (const float*)d_in[1];   // (16, 400, 400, 3)
    const float* poses  = (const float*)d_in[2];   // (16, 4, 4)

    float* feats = (float*)d_out;                  // (P, 16, 3, 5, 5)
    float* coord = feats + FEATS_COUNT;            // (16, P, 2)
    float* Amat  = (float*)d_ws;                   // 48x4 projection rows

    ip_prep<<<1, 16, 0, stream>>>(poses, Amat);
    ip_main<<<NPOINTS / 128, 256, 0, stream>>>(points, images, Amat, feats, coord);
}